// HydraAttention_2516850835590
// MI455X (gfx1250) — compile-verified
//
#include <hip/hip_runtime.h>

// ---------------------------------------------------------------------------
// HydraAttention for MI455X (gfx1250): fp32 WMMA GEMMs + fused linear-attn glue
//   B=4, T=4096, D=1024  ->  M = B*T = 16384
// Global->LDS staging uses GLOBAL_LOAD_ASYNC_TO_LDS_B128 (ASYNCcnt),
// compute uses V_WMMA_F32_16X16X4_F32.
// ---------------------------------------------------------------------------

typedef __attribute__((ext_vector_type(2))) float v2f;
typedef __attribute__((ext_vector_type(8))) float v8f;

#define BATCH   4
#define SEQ     4096
#define DMODEL  1024
#define MTOTAL  (BATCH * SEQ)   // 16384

#define BM 128
#define BN 128
#define BK 16
#define LDA_PAD 20    // 16 + 4 floats  (8B-aligned b64 frag loads, no bank conflicts)
#define LDB_PAD 132   // 128 + 4 floats

// One 16-byte async copy from global address sbase plus offset g_off to LDS address lds_off   (GVS addressing)
__device__ __forceinline__ void async_copy_b128(unsigned int lds_off,
                                                unsigned int g_off,
                                                unsigned long long sbase)
{
    asm volatile("global_load_async_to_lds_b128 %0, %1, %2"
                 :
                 : "v"(lds_off), "v"(g_off), "s"(sbase)
                 : "memory");
}

__device__ __forceinline__ void wait_asynccnt0()
{
    asm volatile("s_wait_asynccnt 0" ::: "memory");
}

__device__ __forceinline__ unsigned int lds_addr32(const void* p)
{
    // generic LDS pointer = {aperture_hi, lds_byte_offset}; low 32 bits are the
    // LDS byte address the async-load VDST operand expects.
    return (unsigned int)(unsigned long long)p;
}

// C[M,N] = A[M,K] @ B[K,N] + bias[N]; all fp32, all dims multiples of tiles.
__global__ __launch_bounds__(256, 1)
void gemm_f32_wmma(const float* __restrict__ A, const float* __restrict__ Bm,
                   const float* __restrict__ bias, float* __restrict__ C,
                   int K, int lda, int ldb, int ldc)
{
    __shared__ float sA[2][BM * LDA_PAD];   // [m][k] tiles
    __shared__ float sB[2][BK * LDB_PAD];   // [k][n] tiles

    const int tid    = threadIdx.x;
    const int lane   = tid & 31;
    const int lane16 = lane & 15;
    const int laneHi = lane >> 4;          // 0 or 1 (wave32 half)
    const int wave   = tid >> 5;           // 8 waves
    const int waveM  = wave & 1;           // 2 waves along M
    const int waveN  = wave >> 1;          // 4 waves along N

    const int blockM = blockIdx.y * BM;
    const int blockN = blockIdx.x * BN;

    // Staging decomposition: per thread, two float4 of A and two float4 of B per BK tile.
    const int aRow0 = (tid)       >> 2, aCol0 = ((tid)       & 3)  << 2;
    const int aRow1 = (tid + 256) >> 2, aCol1 = ((tid + 256) & 3)  << 2;
    const int bRow0 = (tid)       >> 5, bCol0 = ((tid)       & 31) << 2;
    const int bRow1 = (tid + 256) >> 5, bCol1 = ((tid + 256) & 31) << 2;

    // Global element offsets at kk = 0 (fit easily in 32-bit byte offsets).
    const unsigned int baseA0 = (unsigned int)(blockM + aRow0) * lda + aCol0;
    const unsigned int baseA1 = (unsigned int)(blockM + aRow1) * lda + aCol1;
    const unsigned int baseB0 = (unsigned int)bRow0 * ldb + blockN + bCol0;
    const unsigned int baseB1 = (unsigned int)bRow1 * ldb + blockN + bCol1;

    const unsigned long long gA = (unsigned long long)A;
    const unsigned long long gB = (unsigned long long)Bm;

    // LDS byte addresses of this thread's staging slots (buf 0; buf 1 at fixed delta).
    const unsigned int sA0 = lds_addr32(&sA[0][aRow0 * LDA_PAD + aCol0]);
    const unsigned int sA1 = lds_addr32(&sA[0][aRow1 * LDA_PAD + aCol1]);
    const unsigned int sB0 = lds_addr32(&sB[0][bRow0 * LDB_PAD + bCol0]);
    const unsigned int sB1 = lds_addr32(&sB[0][bRow1 * LDB_PAD + bCol1]);
    const unsigned int dA  = (unsigned int)(BM * LDA_PAD * sizeof(float)); // buf stride
    const unsigned int dB  = (unsigned int)(BK * LDB_PAD * sizeof(float));

    auto issue_async = [&](int kk, int buf) {
        const unsigned int ao = (unsigned int)buf * dA;
        const unsigned int bo = (unsigned int)buf * dB;
        async_copy_b128(sA0 + ao, (baseA0 + kk) * 4u, gA);
        async_copy_b128(sA1 + ao, (baseA1 + kk) * 4u, gA);
        async_copy_b128(sB0 + bo, (baseB0 + (unsigned int)kk * ldb) * 4u, gB);
        async_copy_b128(sB1 + bo, (baseB1 + (unsigned int)kk * ldb) * 4u, gB);
    };

    v8f zero = {};
    v8f acc[4][2];
    #pragma unroll
    for (int i = 0; i < 4; ++i)
        #pragma unroll
        for (int j = 0; j < 2; ++j)
            acc[i][j] = zero;

    // One BK-slab of compute: 4 K-steps of 4, 8 wmma per step.
    auto compute = [&](int buf) {
        #pragma unroll
        for (int ks = 0; ks < 4; ++ks) {
            const int k0 = ks * 4 + laneHi * 2;   // ISA 16x4 fp32 A layout: hi half-wave holds K+2
            v2f aF[4], bF[2];
            #pragma unroll
            for (int i = 0; i < 4; ++i) {
                const int row = waveM * 64 + i * 16 + lane16;
                const float* p = &sA[buf][row * LDA_PAD + k0];
                aF[i].x = p[0];
                aF[i].y = p[1];
            }
            #pragma unroll
            for (int j = 0; j < 2; ++j) {
                const int col = waveN * 32 + j * 16 + lane16;
                bF[j].x = sB[buf][(k0)     * LDB_PAD + col];
                bF[j].y = sB[buf][(k0 + 1) * LDB_PAD + col];
            }
            #pragma unroll
            for (int i = 0; i < 4; ++i)
                #pragma unroll
                for (int j = 0; j < 2; ++j)
                    acc[i][j] = __builtin_amdgcn_wmma_f32_16x16x4_f32(
                        false, aF[i], false, bF[j], (short)0, acc[i][j],
                        false, false);
        }
    };

    // Prologue: stream tile 0 straight into LDS.
    issue_async(0, 0);
    wait_asynccnt0();
    __syncthreads();

    int buf = 0;
    for (int kk = BK; kk < K; kk += BK) {
        issue_async(kk, buf ^ 1);      // async-stage next slab (no VGPRs touched)
        compute(buf);                  // WMMA on current slab
        wait_asynccnt0();              // our 4 async copies have landed in LDS
        __syncthreads();               // everyone's copies have landed
        buf ^= 1;
    }
    compute(buf);

    // Epilogue: bias + store.  C/D layout: VGPR r -> rows r (lanes 0-15) / r+8 (lanes 16-31)
    #pragma unroll
    for (int i = 0; i < 4; ++i) {
        #pragma unroll
        for (int j = 0; j < 2; ++j) {
            const int col = blockN + waveN * 32 + j * 16 + lane16;
            const float bv = bias[col];
            const int rbase = blockM + waveM * 64 + i * 16 + 8 * laneHi;
            #pragma unroll
            for (int r = 0; r < 8; ++r) {
                C[(size_t)(rbase + r) * ldc + col] = acc[i][j][r] + bv;
            }
        }
    }
}

// 1/||q_row||, 1/||k_row|| over the feature dim; one block per row of qkv.
__global__ __launch_bounds__(256)
void row_norms(const float* __restrict__ qkv,
               float* __restrict__ inv_q, float* __restrict__ inv_k)
{
    __shared__ float rq[256], rk[256];
    const int m = blockIdx.x;
    const float* row = qkv + (size_t)m * (3 * DMODEL);
    float sq = 0.f, sk = 0.f;
    for (int d = threadIdx.x; d < DMODEL; d += 256) {
        const float q = row[d];
        const float k = row[DMODEL + d];
        sq = fmaf(q, q, sq);
        sk = fmaf(k, k, sk);
    }
    rq[threadIdx.x] = sq;
    rk[threadIdx.x] = sk;
    __syncthreads();
    for (int s = 128; s > 0; s >>= 1) {
        if (threadIdx.x < s) {
            rq[threadIdx.x] += rq[threadIdx.x + s];
            rk[threadIdx.x] += rk[threadIdx.x + s];
        }
        __syncthreads();
    }
    if (threadIdx.x == 0) {
        inv_q[m] = 1.0f / sqrtf(rq[0]);
        inv_k[m] = 1.0f / sqrtf(rk[0]);
    }
}

// Partial kv over 256-timestep chunks: part[chunk][b][d] (deterministic, no atomics).
__global__ __launch_bounds__(256)
void kv_partial(const float* __restrict__ qkv, const float* __restrict__ inv_k,
                float* __restrict__ part)
{
    const int d  = blockIdx.x * 256 + threadIdx.x;  // 0..1023
    const int b  = blockIdx.y;                      // 0..3
    const int tc = blockIdx.z;                      // 0..15
    const int t0 = tc * 256;
    const float* base = qkv + ((size_t)b * SEQ + t0) * (3 * DMODEL);
    const float* ik   = inv_k + b * SEQ + t0;
    float acc = 0.f;
    for (int t = 0; t < 256; ++t) {
        const float* row = base + (size_t)t * (3 * DMODEL);
        acc = fmaf(row[DMODEL + d] * ik[t], row[2 * DMODEL + d], acc);
    }
    part[((size_t)tc * BATCH + b) * DMODEL + d] = acc;
}

// kv[b][d] = sum over 16 chunks in fixed order.
__global__ __launch_bounds__(256)
void kv_final(const float* __restrict__ part, float* __restrict__ kv)
{
    const int i = blockIdx.x * 256 + threadIdx.x;   // 0..(BATCH*DMODEL-1)
    const int b = i >> 10;
    const int d = i & (DMODEL - 1);
    float s = 0.f;
    #pragma unroll
    for (int c = 0; c < 16; ++c)
        s += part[((size_t)c * BATCH + b) * DMODEL + d];
    kv[i] = s;
}

// H = (q * inv_q) * kv, written in place into the q-region of qkv.
__global__ __launch_bounds__(256)
void form_h(float* __restrict__ qkv, const float* __restrict__ inv_q,
            const float* __restrict__ kv)
{
    const size_t idx = (size_t)blockIdx.x * 256 + threadIdx.x;  // MTOTAL*DMODEL
    const int m = (int)(idx >> 10);
    const int d = (int)(idx & (DMODEL - 1));
    const int b = m >> 12;
    float* row = qkv + (size_t)m * (3 * DMODEL);
    row[d] = row[d] * inv_q[m] * kv[b * DMODEL + d];
}

extern "C" void kernel_launch(void* const* d_in, const int* in_sizes, int n_in,
                              void* d_out, int out_size, void* d_ws, size_t ws_size,
                              hipStream_t stream)
{
    const float* x     = (const float*)d_in[0];   // (4,4096,1024)
    const float* W_qkv = (const float*)d_in[1];   // (1024,3072)
    const float* b_qkv = (const float*)d_in[2];   // (3072,)
    const float* W_out = (const float*)d_in[3];   // (1024,1024)
    const float* b_out = (const float*)d_in[4];   // (1024,)
    float* out = (float*)d_out;                   // (4,4096,1024)

    // Workspace layout
    float* qkv    = (float*)d_ws;                          // 16384*3072 f32
    float* inv_q  = qkv + (size_t)MTOTAL * 3 * DMODEL;     // 16384
    float* inv_k  = inv_q + MTOTAL;                        // 16384
    float* kvpart = inv_k + MTOTAL;                        // 16*4*1024
    float* kv     = kvpart + (size_t)16 * BATCH * DMODEL;  // 4*1024

    // 1) qkv = x @ W_qkv + b_qkv
    {
        dim3 grid((3 * DMODEL) / BN, MTOTAL / BM);
        gemm_f32_wmma<<<grid, 256, 0, stream>>>(x, W_qkv, b_qkv, qkv,
                                                DMODEL, DMODEL, 3 * DMODEL, 3 * DMODEL);
    }
    // 2) row norms of q, k
    row_norms<<<MTOTAL, 256, 0, stream>>>(qkv, inv_q, inv_k);
    // 3) kv = sum_t (k/||k||) * v   (split-T partials + fixed-order reduce)
    kv_partial<<<dim3(DMODEL / 256, BATCH, 16), 256, 0, stream>>>(qkv, inv_k, kvpart);
    kv_final<<<(BATCH * DMODEL) / 256, 256, 0, stream>>>(kvpart, kv);
    // 4) H = (q/||q||) * kv  (in place in q-region of qkv)
    form_h<<<(size_t)MTOTAL * DMODEL / 256, 256, 0, stream>>>(qkv, inv_q, kv);
    // 5) out = H @ W_out + b_out   (lda = 3*DMODEL since H lives inside qkv)
    {
        dim3 grid(DMODEL / BN, MTOTAL / BM);
        gemm_f32_wmma<<<grid, 256, 0, stream>>>(qkv, W_out, b_out, out,
                                                DMODEL, 3 * DMODEL, DMODEL, DMODEL);
    }
}